// FocalLoss_37830071943314
// MI455X (gfx1250) — compile-verified
//
#include <hip/hip_runtime.h>

typedef __attribute__((ext_vector_type(4))) float v4f;
typedef __attribute__((ext_vector_type(4))) int   v4i;
typedef __attribute__((ext_vector_type(2))) float v2f;
typedef __attribute__((ext_vector_type(8))) float v8f;

// Focal term via the sigmoid identity. For one-hot ts in {0,1} and logit x,
// with z = (2*ts-1)*x:
//   bce = softplus(-z) = max(-z,0) + log1p(exp(-|z|))
//   pt  = exp(-bce) = sigmoid(z);  1-pt = sigmoid(-z)
// Using u = exp(-|z|), r = rcp(1+u):
//   1-pt = (z>=0) ? u*r : r
// 'at' is 0.25 when ts==1, 0.75 when ts==0.
__device__ __forceinline__ float focal_from_z(float z, float at) {
    float u   = __expf(-fabsf(z));          // u in (0, 1]
    float opu = 1.0f + u;                   // in (1, 2]
    float l   = __logf(opu);                // == log1p(u); exact regime for logf
    float r   = __builtin_amdgcn_rcpf(opu); // v_rcp_f32
    float bce = fmaxf(-z, 0.0f) + l;
    float omp = (z >= 0.0f) ? u * r : r;    // 1 - pt
    return at * omp * omp * bce;            // GAMMA == 2
}

__device__ __forceinline__ float pair_loss(float x0, float x1, int t) {
    // t in {0,1}. Channel 0 has ts=1-t, channel 1 has ts=t.
    float tf  = (float)t;
    float s1  = fmaf(2.0f, tf, -1.0f);        // +1 if t==1 else -1
    float z1  = s1 * x1;                      // (2*ts1-1)*x1
    float z0  = -s1 * x0;                     // (2*ts0-1)*x0
    float at1 = fmaf(-0.5f, tf, 0.75f);       // 0.25 if t==1 else 0.75
    float at0 = fmaf( 0.5f, tf, 0.25f);       // 0.75 if t==1 else 0.25
    return focal_from_z(z0, at0) + focal_from_z(z1, at1);
}

// ---------------------------------------------------------------------------
// Stage 1: streaming partial sums. Bandwidth-bound: 201 MB single pass
// (> 192 MB L2), so B128 loads with non-temporal hints.
// ---------------------------------------------------------------------------
__global__ __launch_bounds__(256) void focal_partial_kernel(
        const float* __restrict__ inp,   // [n_pairs * 2] f32
        const int*   __restrict__ tgt,   // [n_pairs] int32 in {0,1}
        float*       __restrict__ partial,
        long long n_pairs) {
    const long long nchunks = n_pairs >> 2;                       // 4 pairs / chunk
    const long long stride  = (long long)gridDim.x * blockDim.x;
    const long long tid     = (long long)blockIdx.x * blockDim.x + threadIdx.x;

    const v4f* __restrict__ in4 = (const v4f*)inp;   // 2 pairs per v4f
    const v4i* __restrict__ tg4 = (const v4i*)tgt;   // 4 targets per v4i

    float acc = 0.0f;
    for (long long c = tid; c < nchunks; c += stride) {
        v4i t  = __builtin_nontemporal_load(&tg4[c]);
        v4f xa = __builtin_nontemporal_load(&in4[2 * c]);
        v4f xb = __builtin_nontemporal_load(&in4[2 * c + 1]);
        acc += pair_loss(xa.x, xa.y, t.x);
        acc += pair_loss(xa.z, xa.w, t.y);
        acc += pair_loss(xb.x, xb.y, t.z);
        acc += pair_loss(xb.z, xb.w, t.w);
    }
    // scalar tail (n_pairs not a multiple of 4)
    for (long long p = (nchunks << 2) + tid; p < n_pairs; p += stride) {
        acc += pair_loss(inp[2 * p], inp[2 * p + 1], tgt[p]);
    }

    // wave32 shuffle reduction (gfx1250 is wave32-only)
    for (int off = 16; off > 0; off >>= 1)
        acc += __shfl_down(acc, off, 32);

    __shared__ float s[8];
    int lane = threadIdx.x & 31;
    int wave = threadIdx.x >> 5;
    if (lane == 0) s[wave] = acc;
    __syncthreads();
    if (wave == 0) {
        float v = (lane < (int)(blockDim.x >> 5)) ? s[lane] : 0.0f;
        for (int off = 4; off > 0; off >>= 1)
            v += __shfl_down(v, off, 32);
        if (lane == 0) partial[blockIdx.x] = v;
    }
}

// ---------------------------------------------------------------------------
// Stage 2: deterministic final reduction of per-block partials.
// Last 64 -> 1 step uses V_WMMA_F32_16X16X4_F32 with B = ones:
// D[m][n] = sum_k A[m][k]  -> one WMMA yields all 16 row sums.
// ---------------------------------------------------------------------------
__global__ __launch_bounds__(512) void focal_final_kernel(
        const float* __restrict__ partial, int n,
        float* __restrict__ out, float inv_count) {
    __shared__ float s[512];
    const int t = threadIdx.x;

    float v = 0.0f;
    for (int i = t; i < n; i += 512) v += partial[i];
    s[t] = v;
    __syncthreads();
    if (t < 256) s[t] += s[t + 256];
    __syncthreads();
    if (t < 128) s[t] += s[t + 128];
    __syncthreads();
    if (t < 64)  s[t] += s[t + 64];
    __syncthreads();

    // wave 0 (threads 0..31, EXEC all-1s within the wave) does 64 -> 1 via WMMA
    if (t < 32) {
        // 16x4 f32 A layout: lanes 0-15 hold (M=lane, K=0..1), lanes 16-31 hold
        // (M=lane-16, K=2..3). Map s[0..63] so A[m][k] = s[4*m + k].
        int m     = t & 15;
        int kbase = (t < 16) ? 0 : 2;
        v2f a;
        a.x = s[4 * m + kbase];
        a.y = s[4 * m + kbase + 1];
        v2f b = {1.0f, 1.0f};       // B = ones -> D rows replicate row sums of A
        v8f c = {};
        v8f d = __builtin_amdgcn_wmma_f32_16x16x4_f32(
            /*neg_a=*/false, a, /*neg_b=*/false, b,
            /*c_mod=*/(short)0, c, /*reuse_a=*/false, /*reuse_b=*/false);
        // D VGPR r: lanes 0-15 = row r sum, lanes 16-31 = row (8+r) sum.
        float rows = d[0] + d[1] + d[2] + d[3] + d[4] + d[5] + d[6] + d[7];
        float hi   = __shfl(rows, 16, 32);   // rows 8..15 live on lane 16
        if (t == 0) out[0] = (rows + hi) * inv_count;
    }
}

extern "C" void kernel_launch(void* const* d_in, const int* in_sizes, int n_in,
                              void* d_out, int out_size, void* d_ws, size_t ws_size,
                              hipStream_t stream) {
    const float* inp = (const float*)d_in[0];
    const int*   tgt = (const int*)d_in[1];
    const long long n_pairs = (long long)in_sizes[1];   // B*T target count

    float* partial = (float*)d_ws;                      // BLOCKS floats of scratch
    const int BLOCKS = 2048;
    const int TPB    = 256;

    focal_partial_kernel<<<BLOCKS, TPB, 0, stream>>>(inp, tgt, partial, n_pairs);

    const float inv_count = 1.0f / (float)(n_pairs * 2); // mean over [B,T,2]
    focal_final_kernel<<<1, 512, 0, stream>>>(partial, BLOCKS, (float*)d_out, inv_count);
}